// SinkhornLayer_37606733644540
// MI455X (gfx1250) — compile-verified
//
#include <hip/hip_runtime.h>

// Sinkhorn normalization, 512 x (256x256) f32, 21 iterations, T=0.01.
//
// One workgroup (256 threads = 8 wave32) per matrix; matrix resident in LDS
// (stride-258 padding -> conflict-free full-bank ds_load_b64 in BOTH sweeps).
// Log-space Sinkhorn adds only rank-1 (row-const + col-const) offsets, so the
// LDS matrix is READ-ONLY; we iterate only R[256], C[256] (base-2 domain):
//     R_i <- log2 sum_j 2^(la0(i,j)*S - C_j)
//     C_j <- log2 sum_i 2^(la0(i,j)*S - R_i)      (S = 100 * log2(e))
// Output: out(i,j) = 2^(la0(i,j)*S - R_i - C_j).
// Column pass is split: thread t reduces column pair (2*(t&127), +1) over
// rows [128*(t>>7), +128) with float2 loads (full 64-bank utilization);
// halves are merged via small LDS scratch. Transcendentals are single native
// v_exp_f32 / v_log_f32. Initial stage-in uses global_load_async_to_lds_b64.

#define NMAT    256
#define STRIDE  258                     // 258 % 64 == 2 -> conflict-free; rows 8B aligned
#define NITERS  21
#define SCALEF  144.2695040888963407f   // (1/0.01) * log2(e)
#define LDS_FLOATS (NMAT * STRIDE + 6 * NMAT)   // la + R + C + Pm[512] + Ps[512]

#if __has_builtin(__builtin_amdgcn_exp2f)
#define EXP2(x) __builtin_amdgcn_exp2f(x)
#else
#define EXP2(x) exp2f(x)
#endif
#if __has_builtin(__builtin_amdgcn_logf)
#define LOG2(x) __builtin_amdgcn_logf(x)
#else
#define LOG2(x) log2f(x)
#endif

typedef int v2i __attribute__((ext_vector_type(2)));
typedef __attribute__((address_space(1))) v2i* gbl_v2i_p;
typedef __attribute__((address_space(3))) v2i* lds_v2i_p;

extern __shared__ float s_mem[];

__global__ __launch_bounds__(256, 1)
void sinkhorn256_kernel(const float* __restrict__ in, float* __restrict__ out) {
    float* la = s_mem;                        // [256*258] read-only matrix
    float* Rv = s_mem + NMAT * STRIDE;        // [256] row log-offsets (base-2)
    float* Cv = Rv + NMAT;                    // [256] col log-offsets (base-2)
    float* Pm = Cv + NMAT;                    // [512] col partial max (2 halves)
    float* Ps = Pm + 2 * NMAT;                // [512] col partial sum (2 halves)

    const int  t   = threadIdx.x;             // 0..255
    const long b   = blockIdx.x;
    const float* __restrict__ gin  = in  + b * (long)(NMAT * NMAT);
    float* __restrict__       gout = out + b * (long)(NMAT * NMAT);

    // ---- Stage matrix into LDS (coalesced, interleaved mapping). ----------
    // Element e = c*512 + 2t : consecutive lanes -> consecutive 8B -> 256B/wave.
#if __has_builtin(__builtin_amdgcn_global_load_async_to_lds_b64)
    for (int c = 0; c < 128; ++c) {
        const int e = (c << 9) + (t << 1);
        const int l = (e >> 8) * STRIDE + (e & 255);   // 8B-aligned LDS dst
        __builtin_amdgcn_global_load_async_to_lds_b64(
            (gbl_v2i_p)(gin + e),
            (lds_v2i_p)(la + l),
            0, 0);
    }
#if __has_builtin(__builtin_amdgcn_s_wait_asynccnt)
    __builtin_amdgcn_s_wait_asynccnt(0);
#else
    asm volatile("s_wait_asynccnt 0" ::: "memory");
#endif
#else
    for (int c = 0; c < 128; ++c) {
        const int e = (c << 9) + (t << 1);
        const int l = (e >> 8) * STRIDE + (e & 255);
        const float2 v = *(const float2*)(gin + e);
        la[l]     = v.x;
        la[l + 1] = v.y;
    }
#endif
    Cv[t] = 0.0f;
    __syncthreads();

    const float* __restrict__ myrow = la + t * STRIDE;          // row pass
    const int    jp = (t & 127) << 1;                           // col pair base
    const int    h  = t >> 7;                                   // row half
    const int    i0 = h << 7;                                   // 0 or 128
    const float* __restrict__ colbase = la + jp;

    for (int it = 0; it < NITERS; ++it) {
        // ---- Row pass: R[t] = log2 sum_j 2^(la(t,j)*S - C[j]) ------------
        float m = -__builtin_inff();
#pragma unroll 8
        for (int j = 0; j < NMAT; j += 2) {
            const float2 v = *(const float2*)(myrow + j);   // b64, conflict-free
            const float2 c = *(const float2*)(Cv + j);      // b64 broadcast
            const float d0 = __builtin_fmaf(v.x, SCALEF, -c.x);
            const float d1 = __builtin_fmaf(v.y, SCALEF, -c.y);
            m = fmaxf(m, fmaxf(d0, d1));
        }
        float s = 0.0f;
#pragma unroll 8
        for (int j = 0; j < NMAT; j += 2) {
            const float2 v = *(const float2*)(myrow + j);
            const float2 c = *(const float2*)(Cv + j);
            s += EXP2(__builtin_fmaf(v.x, SCALEF, -c.x) - m);
            s += EXP2(__builtin_fmaf(v.y, SCALEF, -c.y) - m);
        }
        Rv[t] = m + LOG2(s);
        __syncthreads();

        // ---- Col pass sweep 1: partial max over rows [i0, i0+128) --------
        float m0 = -__builtin_inff();
        float m1 = -__builtin_inff();
#pragma unroll 8
        for (int i = i0; i < i0 + 128; i += 2) {
            const float2 r  = *(const float2*)(Rv + i);               // broadcast
            const float2 a0 = *(const float2*)(colbase + i * STRIDE); // b64, conflict-free
            const float2 a1 = *(const float2*)(colbase + (i + 1) * STRIDE);
            m0 = fmaxf(m0, fmaxf(__builtin_fmaf(a0.x, SCALEF, -r.x),
                                 __builtin_fmaf(a1.x, SCALEF, -r.y)));
            m1 = fmaxf(m1, fmaxf(__builtin_fmaf(a0.y, SCALEF, -r.x),
                                 __builtin_fmaf(a1.y, SCALEF, -r.y)));
        }
        {
            float2 pm; pm.x = m0; pm.y = m1;
            *(float2*)(Pm + (h << 8) + jp) = pm;
        }
        __syncthreads();

        // ---- Merge halves' maxima (both half-threads compute same value) --
        const float2 pa = *(const float2*)(Pm + jp);          // half 0
        const float2 pb = *(const float2*)(Pm + NMAT + jp);   // half 1
        const float g0 = fmaxf(pa.x, pb.x);
        const float g1 = fmaxf(pa.y, pb.y);

        // ---- Col pass sweep 2: partial sums with global max --------------
        float s0 = 0.0f, s1 = 0.0f;
#pragma unroll 8
        for (int i = i0; i < i0 + 128; i += 2) {
            const float2 r  = *(const float2*)(Rv + i);
            const float2 a0 = *(const float2*)(colbase + i * STRIDE);
            const float2 a1 = *(const float2*)(colbase + (i + 1) * STRIDE);
            s0 += EXP2(__builtin_fmaf(a0.x, SCALEF, -r.x) - g0);
            s0 += EXP2(__builtin_fmaf(a1.x, SCALEF, -r.y) - g0);
            s1 += EXP2(__builtin_fmaf(a0.y, SCALEF, -r.x) - g1);
            s1 += EXP2(__builtin_fmaf(a1.y, SCALEF, -r.y) - g1);
        }
        {
            float2 ps; ps.x = s0; ps.y = s1;
            *(float2*)(Ps + (h << 8) + jp) = ps;
        }
        __syncthreads();

        // ---- Final combine: one column per thread ------------------------
        const float gm = fmaxf(Pm[t], Pm[NMAT + t]);
        const float ss = Ps[t] + Ps[NMAT + t];
        Cv[t] = gm + LOG2(ss);
        __syncthreads();
    }

    // ---- Output: exp(la - R - C) = 2^(la*S - R' - C'), coalesced stores. --
#pragma unroll 4
    for (int c = 0; c < 128; ++c) {
        const int e   = (c << 9) + (t << 1);
        const int row = e >> 8;
        const int col = e & 255;
        const float2 v  = *(const float2*)(la + row * STRIDE + col);
        const float  r  = Rv[row];                       // broadcast
        const float2 cc = *(const float2*)(Cv + col);
        float2 o;
        o.x = EXP2(__builtin_fmaf(v.x, SCALEF, -r) - cc.x);
        o.y = EXP2(__builtin_fmaf(v.y, SCALEF, -r) - cc.y);
        *(float2*)(gout + e) = o;
    }
}

extern "C" void kernel_launch(void* const* d_in, const int* in_sizes, int n_in,
                              void* d_out, int out_size, void* d_ws, size_t ws_size,
                              hipStream_t stream) {
    (void)n_in; (void)out_size; (void)d_ws; (void)ws_size;
    const float* in  = (const float*)d_in[0];
    float*       out = (float*)d_out;

    const int batch = in_sizes[0] / (NMAT * NMAT);          // 512
    const int shmem = (int)(LDS_FLOATS * sizeof(float));    // 270,336 B < 320 KB

    hipError_t err = hipFuncSetAttribute((const void*)&sinkhorn256_kernel,
                                         hipFuncAttributeMaxDynamicSharedMemorySize,
                                         shmem);
    (void)err;
    sinkhorn256_kernel<<<batch, 256, shmem, stream>>>(in, out);
}